// CoalUMLP_85100482003622
// MI455X (gfx1250) — compile-verified
//
#include <hip/hip_runtime.h>
#include <math.h>

// CoalUMLP block for MI455X (gfx1250), wave32, f32 WMMA 16x16x4.
// Pipeline (all GN folded into GEMM weights via prep kernels):
//  k_init   : zero 8 double stat accumulators
//  k_stats  : sum/sumsq of x (GN n1)            [double atomics]
//  k_prep1  : Weff1 = conv1_w * n1_g * rs1, bias fold
//  k_gemm1  : A = Weff1 @ x + beff1, accumulate na1 stats   [2 col-groups/wave]
//  k_gelu_a : A = gelu(gn_na1(A)) in place
//  k_shift3 : S = sum of gelu(conv2{d,h,w}(shift(A))); shifts are predicated
//             B-load offsets (each 16-ch chunk has a uniform shift); na2 stats
//  k_prep3  : Weff3 = conv3_w * na2_g * rs2; alpha/beta fold n1-residual
//  k_gemm3  : X1 = alpha*x + beta + Weff3 @ S; n2 stats     [2 col-groups/wave]
//  k_prep4  : Wefff1 = fc1_w * n2_g * rs3, bias fold
//  k_mlp    : out = X1 + fc2 @ gelu(Wefff1 @ X1 + befff1) + fc2_b
//             448-wide hidden lives in regs/LDS only        [2 col-groups/wave]

#define CH   112
#define NCH  7            // channel tiles of 16
#define NSP  262144       // 64^3 spatial positions (2^18)
#define CNTD ((double)CH * (double)NSP)
#define EPSV 1e-5

typedef __attribute__((ext_vector_type(2))) float v2f;
typedef __attribute__((ext_vector_type(8))) float v8f;

__device__ __forceinline__ v8f wmma_k4(v2f a, v2f b, v8f c) {
  // D = A(16x4 f32) * B(4x16 f32) + C(16x16 f32)
  return __builtin_amdgcn_wmma_f32_16x16x4_f32(
      /*neg_a=*/false, a, /*neg_b=*/false, b,
      /*c_mod=*/(short)0, c, /*reuse_a=*/false, /*reuse_b=*/false);
}

__device__ __forceinline__ v8f vzero8() {
  v8f v = {0.f, 0.f, 0.f, 0.f, 0.f, 0.f, 0.f, 0.f};
  return v;
}

__device__ __forceinline__ float gelu_f(float x) {
  return 0.5f * x * (1.0f + erff(x * 0.7071067811865475f));
}

__device__ __forceinline__ float wave_sum(float v) {
#pragma unroll
  for (int off = 16; off > 0; off >>= 1) v += __shfl_xor(v, off, 32);
  return v;
}

// ---------------------------------------------------------------- stats init
__global__ void k_init(double* stats) {
  if (threadIdx.x < 8) stats[threadIdx.x] = 0.0;
}

// ------------------------------------------------------------- GN stats of x
__global__ void k_stats(const float* __restrict__ x, double* __restrict__ stats) {
  long long i = (long long)blockIdx.x * blockDim.x + threadIdx.x;
  long long stride = (long long)gridDim.x * blockDim.x;
  float s = 0.f, q = 0.f;
  for (long long j = i; j < (long long)CH * NSP; j += stride) {
    float v = x[j];
    s += v; q += v * v;
  }
  s = wave_sum(s); q = wave_sum(q);
  if ((threadIdx.x & 31) == 0) {
    atomicAdd(&stats[0], (double)s);
    atomicAdd(&stats[1], (double)q);
  }
}

// ---------------------------------------------- fold gn(n1) into conv1 weights
__global__ void k_prep1(const float* __restrict__ w1, const float* __restrict__ b1,
                        const float* __restrict__ g, const float* __restrict__ bn,
                        const double* __restrict__ stats,
                        float* __restrict__ weff, float* __restrict__ beff) {
  double mu = stats[0] / CNTD;
  double var = stats[1] / CNTD - mu * mu;
  float rs = (float)(1.0 / sqrt(var + EPSV));
  float muf = (float)mu;
  for (int o = threadIdx.x; o < CH; o += blockDim.x) {
    float acc = b1[o];
    for (int c = 0; c < CH; c++) {
      float w = w1[o * CH + c];
      weff[o * CH + c] = w * g[c] * rs;
      acc += w * (bn[c] - g[c] * rs * muf);
    }
    beff[o] = acc;
  }
}

// ---------- GEMM: A = Weff1 @ x + beff1, na1 stats; 2 column groups per wave
#define WAVES 8
#define GT 2
__global__ __launch_bounds__(256, 1)
void k_gemm1(const float* __restrict__ x, const float* __restrict__ weff,
             const float* __restrict__ beff, float* __restrict__ outA,
             double* __restrict__ stats) {
  __shared__ float sW[CH * CH];
  __shared__ float sB[CH];
  for (int i = threadIdx.x; i < CH * CH; i += blockDim.x) sW[i] = weff[i];
  for (int i = threadIdx.x; i < CH; i += blockDim.x) sB[i] = beff[i];
  __syncthreads();

  int lane = threadIdx.x & 31;
  int wid = threadIdx.x >> 5;
  int n0 = (blockIdx.x * WAVES + wid) * (16 * GT);
  int nl = lane & 15;
  int khalf = (lane >> 4) << 1;        // A/B K-pair select per half-wave
  int rbase = (lane >> 4) ? 8 : 0;     // C/D row base per half-wave

  v8f acc[GT][NCH];
#pragma unroll
  for (int g = 0; g < GT; g++)
#pragma unroll
    for (int t = 0; t < NCH; t++) acc[g][t] = vzero8();

  for (int kc = 0; kc < 28; kc++) {
    int crow = kc * 4 + khalf;
    v2f b[GT];
#pragma unroll
    for (int g = 0; g < GT; g++) {
      long long base = (long long)crow * NSP + n0 + g * 16 + nl;
      b[g].x = x[base];
      b[g].y = x[base + NSP];
    }
#pragma unroll
    for (int t = 0; t < NCH; t++) {
      const float* q = &sW[(t * 16 + nl) * CH + crow];
      v2f a; a.x = q[0]; a.y = q[1];
#pragma unroll
      for (int g = 0; g < GT; g++) acc[g][t] = wmma_k4(a, b[g], acc[g][t]);
    }
  }

  float s = 0.f, q = 0.f;
#pragma unroll
  for (int g = 0; g < GT; g++)
#pragma unroll
    for (int t = 0; t < NCH; t++)
#pragma unroll
      for (int r = 0; r < 8; r++) {
        int row = t * 16 + rbase + r;
        float v = acc[g][t][r] + sB[row];
        outA[(long long)row * NSP + n0 + g * 16 + nl] = v;
        s += v; q += v * v;
      }
  s = wave_sum(s); q = wave_sum(q);
  if (lane == 0) {
    atomicAdd(&stats[2], (double)s);
    atomicAdd(&stats[3], (double)q);
  }
}

// ------------------------------------------- A = gelu(gn_na1(A)) elementwise
__global__ void k_gelu_a(float* __restrict__ A, const double* __restrict__ stats,
                         const float* __restrict__ g, const float* __restrict__ bn) {
  double mu = stats[2] / CNTD;
  double var = stats[3] / CNTD - mu * mu;
  float rs = (float)(1.0 / sqrt(var + EPSV));
  float muf = (float)mu;
  long long i = (long long)blockIdx.x * blockDim.x + threadIdx.x;
  long long stride = (long long)gridDim.x * blockDim.x;
  for (long long j = i; j < (long long)CH * NSP; j += stride) {
    int c = (int)(j >> 18);            // NSP == 2^18
    float v = (A[j] - muf) * rs * g[c] + bn[c];
    A[j] = gelu_f(v);
  }
}

// ----- 3 shifted GEMM branches fused; shifts become predicated B-load offsets
__global__ __launch_bounds__(256, 1)
void k_shift3(const float* __restrict__ A,
              const float* __restrict__ w_d, const float* __restrict__ b_d,
              const float* __restrict__ w_h, const float* __restrict__ b_h,
              const float* __restrict__ w_w, const float* __restrict__ b_w,
              float* __restrict__ S, double* __restrict__ stats) {
  __shared__ float sW[CH * CH];
  __shared__ float sB[CH];
  int lane = threadIdx.x & 31;
  int wid = threadIdx.x >> 5;
  int n0 = (blockIdx.x * WAVES + wid) * 16;
  int col = n0 + (lane & 15);
  int khalf = (lane >> 4) << 1;
  int rbase = (lane >> 4) ? 8 : 0;
  int d = col >> 12, h = (col >> 6) & 63, w = col & 63;

  v8f Sacc[NCH];
#pragma unroll
  for (int t = 0; t < NCH; t++) Sacc[t] = vzero8();

  const float* wsrc[3] = {w_d, w_h, w_w};
  const float* bsrc[3] = {b_d, b_h, b_w};

  for (int br = 0; br < 3; br++) {
    __syncthreads();   // prior branch done with sW
    for (int i = threadIdx.x; i < CH * CH; i += blockDim.x) sW[i] = wsrc[br][i];
    for (int i = threadIdx.x; i < CH; i += blockDim.x) sB[i] = bsrc[br][i];
    __syncthreads();

    v8f acc[NCH];
#pragma unroll
    for (int t = 0; t < NCH; t++) acc[t] = vzero8();

    for (int kc = 0; kc < 28; kc++) {
      int ct = kc >> 2;                // 16-channel chunk == shift group
      int sh = ct - 3;
      int dd, hh, ww;
      if (br == 0)      { dd = d - sh;  hh = h;           ww = w; }
      else if (br == 1) { dd = d - 3;   hh = h - sh - 3;  ww = w - 3; }
      else              { dd = d - 6;   hh = h - 6;       ww = w - sh - 6; }
      bool ok = ((unsigned)dd < 64u) && ((unsigned)hh < 64u) && ((unsigned)ww < 64u);
      long long sp = ((long long)dd << 12) + (hh << 6) + ww;
      int crow = kc * 4 + khalf;
      v2f b;
      b.x = ok ? A[(long long)crow * NSP + sp] : 0.f;
      b.y = ok ? A[(long long)(crow + 1) * NSP + sp] : 0.f;
#pragma unroll
      for (int t = 0; t < NCH; t++) {
        const float* qq = &sW[(t * 16 + (lane & 15)) * CH + crow];
        v2f a; a.x = qq[0]; a.y = qq[1];
        acc[t] = wmma_k4(a, b, acc[t]);
      }
    }
#pragma unroll
    for (int t = 0; t < NCH; t++) {
#pragma unroll
      for (int r = 0; r < 8; r++) {
        int row = t * 16 + rbase + r;
        Sacc[t][r] += gelu_f(acc[t][r] + sB[row]);
      }
    }
  }

  float s = 0.f, q = 0.f;
#pragma unroll
  for (int t = 0; t < NCH; t++) {
#pragma unroll
    for (int r = 0; r < 8; r++) {
      int row = t * 16 + rbase + r;
      float v = Sacc[t][r];
      S[(long long)row * NSP + col] = v;
      s += v; q += v * v;
    }
  }
  s = wave_sum(s); q = wave_sum(q);
  if (lane == 0) {
    atomicAdd(&stats[4], (double)s);
    atomicAdd(&stats[5], (double)q);
  }
}

// ------------------ fold gn(na2) into conv3; fold h0 residual into alpha/beta
__global__ void k_prep3(const float* __restrict__ w3, const float* __restrict__ b3,
                        const float* __restrict__ g2, const float* __restrict__ bn2,
                        const float* __restrict__ g1, const float* __restrict__ bn1,
                        const double* __restrict__ stats, float* __restrict__ weff,
                        float* __restrict__ alpha, float* __restrict__ beta) {
  double mu1 = stats[0] / CNTD, var1 = stats[1] / CNTD - mu1 * mu1;
  float rs1 = (float)(1.0 / sqrt(var1 + EPSV)), mu1f = (float)mu1;
  double mu2 = stats[4] / CNTD, var2 = stats[5] / CNTD - mu2 * mu2;
  float rs2 = (float)(1.0 / sqrt(var2 + EPSV)), mu2f = (float)mu2;
  for (int o = threadIdx.x; o < CH; o += blockDim.x) {
    float acc = b3[o];
    for (int c = 0; c < CH; c++) {
      float wv = w3[o * CH + c];
      weff[o * CH + c] = wv * g2[c] * rs2;
      acc += wv * (bn2[c] - g2[c] * rs2 * mu2f);
    }
    alpha[o] = 1.f + g1[o] * rs1;                  // x + h0 row scale
    beta[o] = bn1[o] - g1[o] * rs1 * mu1f + acc;   // h0 offset + conv3 bias fold
  }
}

// ------ X1 = alpha*x + beta + Weff3 @ S; n2 stats; 2 column groups per wave
__global__ __launch_bounds__(256, 1)
void k_gemm3(const float* __restrict__ Sin, const float* __restrict__ x,
             const float* __restrict__ weff, const float* __restrict__ alpha,
             const float* __restrict__ beta, float* __restrict__ X1,
             double* __restrict__ stats) {
  __shared__ float sW[CH * CH];
  __shared__ float sAl[CH];
  __shared__ float sBe[CH];
  for (int i = threadIdx.x; i < CH * CH; i += blockDim.x) sW[i] = weff[i];
  for (int i = threadIdx.x; i < CH; i += blockDim.x) { sAl[i] = alpha[i]; sBe[i] = beta[i]; }
  __syncthreads();

  int lane = threadIdx.x & 31;
  int wid = threadIdx.x >> 5;
  int n0 = (blockIdx.x * WAVES + wid) * (16 * GT);
  int nl = lane & 15;
  int khalf = (lane >> 4) << 1;
  int rbase = (lane >> 4) ? 8 : 0;

  v8f acc[GT][NCH];
#pragma unroll
  for (int g = 0; g < GT; g++)
#pragma unroll
    for (int t = 0; t < NCH; t++) acc[g][t] = vzero8();

  for (int kc = 0; kc < 28; kc++) {
    int crow = kc * 4 + khalf;
    v2f b[GT];
#pragma unroll
    for (int g = 0; g < GT; g++) {
      long long base = (long long)crow * NSP + n0 + g * 16 + nl;
      b[g].x = Sin[base];
      b[g].y = Sin[base + NSP];
    }
#pragma unroll
    for (int t = 0; t < NCH; t++) {
      const float* q = &sW[(t * 16 + nl) * CH + crow];
      v2f a; a.x = q[0]; a.y = q[1];
#pragma unroll
      for (int g = 0; g < GT; g++) acc[g][t] = wmma_k4(a, b[g], acc[g][t]);
    }
  }

  float s = 0.f, q = 0.f;
#pragma unroll
  for (int g = 0; g < GT; g++)
#pragma unroll
    for (int t = 0; t < NCH; t++)
#pragma unroll
      for (int r = 0; r < 8; r++) {
        int row = t * 16 + rbase + r;
        long long idx = (long long)row * NSP + n0 + g * 16 + nl;
        float v = acc[g][t][r] + sAl[row] * x[idx] + sBe[row];
        X1[idx] = v;
        s += v; q += v * v;
      }
  s = wave_sum(s); q = wave_sum(q);
  if (lane == 0) {
    atomicAdd(&stats[6], (double)s);
    atomicAdd(&stats[7], (double)q);
  }
}

// ------------------ fold gn(n2) into fc1 weights
__global__ void k_prep4(const float* __restrict__ wf1, const float* __restrict__ bf1,
                        const float* __restrict__ g3, const float* __restrict__ bn3,
                        const double* __restrict__ stats,
                        float* __restrict__ weff, float* __restrict__ beff) {
  double mu = stats[6] / CNTD, var = stats[7] / CNTD - mu * mu;
  float rs = (float)(1.0 / sqrt(var + EPSV)), muf = (float)mu;
  for (int j = threadIdx.x; j < 4 * CH; j += blockDim.x) {
    float acc = bf1[j];
    for (int c = 0; c < CH; c++) {
      float wv = wf1[j * CH + c];
      weff[j * CH + c] = wv * g3[c] * rs;
      acc += wv * (bn3[c] - g3[c] * rs * muf);
    }
    beff[j] = acc;
  }
}

// --- MLP: out = X1 + fc2 @ gelu(Wefff1 @ X1 + befff1) + fc2_b
//     2 column groups per wave: every weight fetch (A operand) feeds 2 WMMAs.
#define MWAVES 4
#define MGT 2
__global__ __launch_bounds__(128, 1)
void k_mlp(const float* __restrict__ X1, const float* __restrict__ wf1,
           const float* __restrict__ bf1, const float* __restrict__ wf2,
           const float* __restrict__ bf2, float* __restrict__ out) {
  __shared__ float sX[MWAVES * MGT * 16 * CH];   // per-wave x1 tiles, [g][n][c]
  __shared__ float sH[MWAVES * MGT * 16 * 16];   // per-wave hidden tiles, [g][n][j]
  int lane = threadIdx.x & 31;
  int wid = threadIdx.x >> 5;
  int n0 = (blockIdx.x * MWAVES + wid) * (16 * MGT);
  float* wX = &sX[wid * MGT * 16 * CH];
  float* wH = &sH[wid * MGT * 16 * 16];

  for (int i = lane; i < MGT * 16 * CH; i += 32) {
    int g = i / (16 * CH);
    int rem = i - g * (16 * CH);
    int cch = rem >> 4, nn = rem & 15;
    wX[g * (16 * CH) + nn * CH + cch] = X1[(long long)cch * NSP + n0 + g * 16 + nn];
  }
  __syncthreads();

  int nl = lane & 15;
  int khalf = (lane >> 4) << 1;
  int rbase = (lane >> 4) ? 8 : 0;

  v8f oacc[MGT][NCH];
#pragma unroll
  for (int g = 0; g < MGT; g++)
#pragma unroll
    for (int t = 0; t < NCH; t++) oacc[g][t] = vzero8();

  for (int jt = 0; jt < 28; jt++) {
    // hidden tile jt: 16 x (16*MGT) over K=112
    v8f hacc[MGT];
#pragma unroll
    for (int g = 0; g < MGT; g++) hacc[g] = vzero8();
    for (int kc = 0; kc < 28; kc++) {
      int crow = kc * 4 + khalf;
      const float* qa = &wf1[(jt * 16 + nl) * CH + crow];
      v2f a; a.x = qa[0]; a.y = qa[1];
#pragma unroll
      for (int g = 0; g < MGT; g++) {
        const float* qb = &wX[g * (16 * CH) + nl * CH + crow];
        v2f b; b.x = qb[0]; b.y = qb[1];
        hacc[g] = wmma_k4(a, b, hacc[g]);
      }
    }
    __syncthreads();   // previous jt's wH fully consumed (uniform trip count)
#pragma unroll
    for (int g = 0; g < MGT; g++)
#pragma unroll
      for (int r = 0; r < 8; r++) {
        int jrow = jt * 16 + rbase + r;
        wH[g * 256 + nl * 16 + rbase + r] = gelu_f(hacc[g][r] + bf1[jrow]);
      }
    __syncthreads();
    // out += fc2[:, jt*16 : jt*16+16] @ hid
#pragma unroll
    for (int kc = 0; kc < 4; kc++) {
      int jj = kc * 4 + khalf;
      v2f b[MGT];
#pragma unroll
      for (int g = 0; g < MGT; g++) {
        const float* qb = &wH[g * 256 + nl * 16 + jj];
        b[g].x = qb[0]; b[g].y = qb[1];
      }
#pragma unroll
      for (int t = 0; t < NCH; t++) {
        const float* qa = &wf2[(t * 16 + nl) * (4 * CH) + jt * 16 + jj];
        v2f a; a.x = qa[0]; a.y = qa[1];
#pragma unroll
        for (int g = 0; g < MGT; g++) oacc[g][t] = wmma_k4(a, b[g], oacc[g][t]);
      }
    }
  }
#pragma unroll
  for (int g = 0; g < MGT; g++)
#pragma unroll
    for (int t = 0; t < NCH; t++)
#pragma unroll
      for (int r = 0; r < 8; r++) {
        int row = t * 16 + rbase + r;
        out[(long long)row * NSP + n0 + g * 16 + nl] =
            oacc[g][t][r] + wX[g * (16 * CH) + nl * CH + row] + bf2[row];
      }
}

// ------------------------------------------------------------- host launcher
static const long long OFF_W1 = 16;            // after 8 doubles of stats
static const long long OFF_B1 = OFF_W1 + 112 * 112;
static const long long OFF_W3 = OFF_B1 + 112;
static const long long OFF_AL = OFF_W3 + 112 * 112;
static const long long OFF_BE = OFF_AL + 112;
static const long long OFF_WF = OFF_BE + 112;
static const long long OFF_BF = OFF_WF + 448 * 112;
static const long long OFF_A  = 76288;                       // 128-aligned
static const long long OFF_S  = OFF_A + (long long)112 * 262144;
static const long long OFF_X1 = OFF_S + (long long)112 * 262144;

extern "C" void kernel_launch(void* const* d_in, const int* in_sizes, int n_in,
                              void* d_out, int out_size, void* d_ws, size_t ws_size,
                              hipStream_t stream) {
  (void)in_sizes; (void)n_in; (void)out_size; (void)ws_size;
  const float* x       = (const float*)d_in[0];
  const float* conv1_w = (const float*)d_in[1];
  const float* conv1_b = (const float*)d_in[2];
  const float* conv2d_w = (const float*)d_in[3];
  const float* conv2d_b = (const float*)d_in[4];
  const float* conv2h_w = (const float*)d_in[5];
  const float* conv2h_b = (const float*)d_in[6];
  const float* conv2w_w = (const float*)d_in[7];
  const float* conv2w_b = (const float*)d_in[8];
  const float* conv3_w = (const float*)d_in[9];
  const float* conv3_b = (const float*)d_in[10];
  const float* fc1_w = (const float*)d_in[11];
  const float* fc1_b = (const float*)d_in[12];
  const float* fc2_w = (const float*)d_in[13];
  const float* fc2_b = (const float*)d_in[14];
  const float* n1_g = (const float*)d_in[15];
  const float* n1_b = (const float*)d_in[16];
  const float* na1_g = (const float*)d_in[17];
  const float* na1_b = (const float*)d_in[18];
  const float* na2_g = (const float*)d_in[19];
  const float* na2_b = (const float*)d_in[20];
  const float* n2_g = (const float*)d_in[21];
  const float* n2_b = (const float*)d_in[22];

  float* out = (float*)d_out;
  double* stats = (double*)d_ws;
  float* wsf = (float*)d_ws;
  float* weff1 = wsf + OFF_W1; float* beff1 = wsf + OFF_B1;
  float* weff3 = wsf + OFF_W3; float* alpha = wsf + OFF_AL; float* beta = wsf + OFF_BE;
  float* wefff = wsf + OFF_WF; float* befff = wsf + OFF_BF;
  float* A = wsf + OFF_A; float* S = wsf + OFF_S; float* X1 = wsf + OFF_X1;

  hipLaunchKernelGGL(k_init, dim3(1), dim3(32), 0, stream, stats);
  hipLaunchKernelGGL(k_stats, dim3(2048), dim3(256), 0, stream, x, stats);
  hipLaunchKernelGGL(k_prep1, dim3(1), dim3(128), 0, stream,
                     conv1_w, conv1_b, n1_g, n1_b, stats, weff1, beff1);
  // 16384 column groups / (8 waves * 2 groups) = 1024 blocks
  hipLaunchKernelGGL(k_gemm1, dim3(1024), dim3(256), 0, stream,
                     x, weff1, beff1, A, stats);
  hipLaunchKernelGGL(k_gelu_a, dim3(4096), dim3(256), 0, stream,
                     A, stats, na1_g, na1_b);
  // 16384 / 8 = 2048 blocks (single group per wave: register pressure)
  hipLaunchKernelGGL(k_shift3, dim3(2048), dim3(256), 0, stream,
                     A, conv2d_w, conv2d_b, conv2h_w, conv2h_b, conv2w_w, conv2w_b,
                     S, stats);
  hipLaunchKernelGGL(k_prep3, dim3(1), dim3(128), 0, stream,
                     conv3_w, conv3_b, na2_g, na2_b, n1_g, n1_b, stats,
                     weff3, alpha, beta);
  hipLaunchKernelGGL(k_gemm3, dim3(1024), dim3(256), 0, stream,
                     S, x, weff3, alpha, beta, X1, stats);
  hipLaunchKernelGGL(k_prep4, dim3(1), dim3(128), 0, stream,
                     fc1_w, fc1_b, n2_g, n2_b, stats, wefff, befff);
  // 16384 / (4 waves * 2 groups) = 2048 blocks
  hipLaunchKernelGGL(k_mlp, dim3(2048), dim3(128), 0, stream,
                     X1, wefff, befff, fc2_w, fc2_b, out);
}